// ClassBasedDecoder_71399536328749
// MI455X (gfx1250) — compile-verified
//
#include <hip/hip_runtime.h>
#include <hip/hip_bf16.h>

#define NHID  512
#define NCLS  250
#define CHUNK 200
#define NTOK  2048
#define OUTW  (NCLS + CHUNK)   // 450

typedef __attribute__((ext_vector_type(2))) float v2f;
typedef __attribute__((ext_vector_type(4))) float v4f;
typedef __attribute__((ext_vector_type(8))) float v8f;

// ---------------------------------------------------------------------------
// Token grouping: bucket tokens by class so each Ww[c] is streamed once.
// ws ints: counts[256] | offsets[256] | cursor[256] | sorted[2048]
// ---------------------------------------------------------------------------
__global__ void k_init(int* __restrict__ counts, int* __restrict__ cursor) {
    int i = threadIdx.x;
    if (i < 256) { counts[i] = 0; cursor[i] = 0; }
}

__global__ void k_count(const int* __restrict__ cls, int* __restrict__ counts) {
    int n = blockIdx.x * blockDim.x + threadIdx.x;
    if (n < NTOK) atomicAdd(&counts[cls[n]], 1);
}

__global__ void k_scan(const int* __restrict__ counts, int* __restrict__ offsets) {
    if (threadIdx.x == 0) {
        int s = 0;
        for (int c = 0; c < NCLS; ++c) { offsets[c] = s; s += counts[c]; }
        offsets[NCLS] = s;
    }
}

__global__ void k_scatter(const int* __restrict__ cls, const int* __restrict__ offsets,
                          int* __restrict__ cursor, int* __restrict__ sorted) {
    int n = blockIdx.x * blockDim.x + threadIdx.x;
    if (n < NTOK) {
        int c = cls[n];
        int p = atomicAdd(&cursor[c], 1);
        sorted[offsets[c] + p] = n;
    }
}

// ---------------------------------------------------------------------------
// Shared f32 WMMA micro-kernel: D[16,16] += A[16,512] x B[512,16].
// K remapped so each lane does one b128 load per 8 K-values, feeding two
// v_wmma_f32_16x16x4_f32. A and B use the identical K mapping -> exact sum.
// ---------------------------------------------------------------------------
__device__ __forceinline__ v8f wmma_row_dot(const float* __restrict__ ap,
                                            const float* __restrict__ bp) {
    v8f acc = {};
#pragma unroll 8
    for (int k = 0; k < NHID; k += 8) {
        v4f a = *(const v4f*)(ap + k);
        v4f b = *(const v4f*)(bp + k);
        v2f a0 = {a.x, a.y}, a1 = {a.z, a.w};
        v2f b0 = {b.x, b.y}, b1 = {b.z, b.w};
        acc = __builtin_amdgcn_wmma_f32_16x16x4_f32(false, a0, false, b0,
                                                    (short)0, acc, false, false);
        acc = __builtin_amdgcn_wmma_f32_16x16x4_f32(false, a1, false, b1,
                                                    (short)0, acc, false, false);
    }
    return acc;
}

// ---------------------------------------------------------------------------
// p_class: out[n, 0:250] = x @ Wc^T + bc.  One wave per 16x16 tile.
// grid = (ceil(NCLS/16), NTOK/16), block = 32.
// ---------------------------------------------------------------------------
__global__ __launch_bounds__(32) void k_class(const float* __restrict__ x,
                                              const float* __restrict__ Wc,
                                              const float* __restrict__ bc,
                                              float* __restrict__ out) {
    const int lane = threadIdx.x & 31;
    const int lo = lane & 15;        // A: M row / B: N col
    const int hi = lane >> 4;        // K sub-offset
    const int m0 = blockIdx.y << 4;  // token base
    const int n0 = blockIdx.x << 4;  // class base

    const int arow = m0 + lo;                            // always in range
    const int brow_raw = n0 + lo;
    const int brow = brow_raw < NCLS ? brow_raw : NCLS - 1;  // clamp reads

    const float* ap = x  + (size_t)arow * NHID + 4 * hi;
    const float* bp = Wc + (size_t)brow * NHID + 4 * hi;

    v8f acc = wmma_row_dot(ap, bp);

    if (brow_raw < NCLS) {
        const float bias = bc[brow_raw];
#pragma unroll
        for (int r = 0; r < 8; ++r) {
            const int tok = m0 + r + 8 * hi;             // C layout: M = r + 8*hi
            out[(size_t)tok * OUTW + brow_raw] = acc[r] + bias;
        }
    }
}

// ---------------------------------------------------------------------------
// p_words: tokens grouped by class; out[n, 250:450] = Ww[c] @ x[n] + bw[c].
// grid = (ceil(CHUNK/16), NCLS), block = 32. Each wave: one 16-word tile of
// one class; loops over 16-token groups (usually one). Ww streamed once.
// ---------------------------------------------------------------------------
__global__ __launch_bounds__(32) void k_words(const float* __restrict__ x,
                                              const float* __restrict__ Ww,
                                              const float* __restrict__ bw,
                                              const int* __restrict__ offsets,
                                              const int* __restrict__ sorted,
                                              float* __restrict__ out) {
    const int c  = blockIdx.y;
    const int w0 = blockIdx.x << 4;       // word base within chunk
    const int lane = threadIdx.x & 31;
    const int lo = lane & 15;
    const int hi = lane >> 4;

    const int beg = offsets[c];
    const int cnt = offsets[c + 1] - beg;
    if (cnt == 0) return;                 // wave-uniform: EXEC stays all-ones

    const int wrow_raw = w0 + lo;
    const int wrow = wrow_raw < CHUNK ? wrow_raw : CHUNK - 1;   // clamp reads
    const float* bp = Ww + ((size_t)c * CHUNK + wrow) * NHID + 4 * hi;
    const float bias = bw[c * CHUNK + wrow];

    for (int t0 = 0; t0 < cnt; t0 += 16) {
        const int ti  = t0 + lo;
        const int tok = sorted[beg + (ti < cnt ? ti : cnt - 1)]; // clamp gather
        const float* ap = x + (size_t)tok * NHID + 4 * hi;

        v8f acc = wmma_row_dot(ap, bp);

        if (wrow_raw < CHUNK) {
#pragma unroll
            for (int r = 0; r < 8; ++r) {
                const int tI = t0 + r + 8 * hi;
                if (tI < cnt) {
                    const int tk = sorted[beg + tI];
                    out[(size_t)tk * OUTW + NCLS + wrow_raw] = acc[r] + bias;
                }
            }
        }
    }
}

// ---------------------------------------------------------------------------
extern "C" void kernel_launch(void* const* d_in, const int* in_sizes, int n_in,
                              void* d_out, int out_size, void* d_ws, size_t ws_size,
                              hipStream_t stream) {
    const float* x   = (const float*)d_in[0];
    const float* Wc  = (const float*)d_in[1];
    const float* bc  = (const float*)d_in[2];
    const float* Ww  = (const float*)d_in[3];
    const float* bw  = (const float*)d_in[4];
    const int*   cls = (const int*)d_in[5];
    float* out = (float*)d_out;

    int* counts  = (int*)d_ws;
    int* offsets = counts + 256;
    int* cursor  = offsets + 256;
    int* sorted  = cursor + 256;

    k_init<<<1, 256, 0, stream>>>(counts, cursor);
    k_count<<<(NTOK + 255) / 256, 256, 0, stream>>>(cls, counts);
    k_scan<<<1, 32, 0, stream>>>(counts, offsets);
    k_scatter<<<(NTOK + 255) / 256, 256, 0, stream>>>(cls, offsets, cursor, sorted);

    k_class<<<dim3((NCLS + 15) / 16, NTOK / 16), 32, 0, stream>>>(x, Wc, bc, out);
    k_words<<<dim3((CHUNK + 15) / 16, NCLS), 32, 0, stream>>>(x, Ww, bw, offsets, sorted, out);
}